// EMDLossAlt_31980326486600
// MI455X (gfx1250) — compile-verified
//
#include <hip/hip_runtime.h>
#include <math.h>

// Debiased Sinkhorn divergence, geomloss-style, for B=8 pairs of 2048x3 clouds.
// eps = blur^p = 0.05^2 = 0.0025, 50 log-domain iterations, uniform weights.
//
// MI455X strategy:
//  - Rebuild cost tiles with V_WMMA_F32_16X16X4_F32. The free K=4 slot carries
//    the column term (A.k3 = 1, B.k3 = (g_j - 0.5|y_j|^2)/eps*log2e) and the
//    WMMA C operand carries the per-row term, so the WMMA output IS the
//    finished logit in log2 units: no per-tile VALU arithmetic at all.
//  - Two-pass logsumexp (max sweep with v_max3, then exp2 sweep with the max
//    folded into C): 1 native v_exp_f32 per matrix element.
//  - B fragments pre-swizzled in LDS to the exact WMMA lane layout
//    (ds_load_b64 per tile, no cndmasks); xy halves staged via
//    GLOBAL_LOAD_ASYNC_TO_LDS_B64 when available.
//  - One launch per half Sinkhorn iteration (100 graph nodes) so the whole
//    device works on all 24 OT problems every pass.

typedef __attribute__((ext_vector_type(2))) float v2f;
typedef __attribute__((ext_vector_type(8))) float v8f;
typedef __attribute__((ext_vector_type(2))) int   v2i;
typedef __attribute__((address_space(1))) v2i*    gbl_v2i_p;  // global (AS1)
typedef __attribute__((address_space(3))) v2i*    lds_v2i_p;  // LDS (AS3)

#define SK_EPS     0.0025f
#define NPTS       2048
#define NBATCH     8
#define NPROB      24            // 8 batches x {xy, xx, yy}
#define LOG_N      7.6246189861593985f   // log(2048)
#define LOG2E      1.4426950408889634f
#define LN2        0.6931471805599453f
#define SK_SCALE   (400.0f * LOG2E)      // 1/eps * log2(e)
#define SK_ITERS   50
#define BLK        256           // 8 waves, each owns a 16-row tile
#define ROWS_PER_BLOCK 128
#define BLOCKS_PER_PROB (NPTS / ROWS_PER_BLOCK)   // 16
#define NTILES     (NPTS / 16)   // 128 column tiles

// ---------------------------------------------------------------------------
// Init: pad points to float4, precompute |p|^2.
// ---------------------------------------------------------------------------
__global__ void sk_init(const float* __restrict__ pcs1,
                        const float* __restrict__ pcs2,
                        float4* __restrict__ xpad, float4* __restrict__ ypad,
                        float* __restrict__ xsq, float* __restrict__ ysq) {
  int idx = blockIdx.x * blockDim.x + threadIdx.x;   // 0 .. 8*2048-1
  if (idx >= NBATCH * NPTS) return;
  float a0 = pcs1[idx * 3 + 0], a1 = pcs1[idx * 3 + 1], a2 = pcs1[idx * 3 + 2];
  float b0 = pcs2[idx * 3 + 0], b1 = pcs2[idx * 3 + 1], b2 = pcs2[idx * 3 + 2];
  xpad[idx] = make_float4(a0, a1, a2, 0.0f);
  ypad[idx] = make_float4(b0, b1, b2, 0.0f);
  xsq[idx] = a0 * a0 + a1 * a1 + a2 * a2;
  ysq[idx] = b0 * b0 + b1 * b1 + b2 * b2;
}

// ---------------------------------------------------------------------------
// One half Sinkhorn iteration for all 24 OT problems.
// updateF=1: f_i = -eps*(LSE_j((g_j - C_ij)/eps) - logM)   (reduce over cols=Y)
// updateF=0: g_j = -eps*(LSE_i((f_i - C_ij)/eps) - logN)   (reduce over cols=X)
// Cost is symmetric, so both halves are the same kernel with rows/cols swapped.
// ---------------------------------------------------------------------------
__launch_bounds__(BLK)
__global__ void sk_pass(const float4* __restrict__ xpad,
                        const float4* __restrict__ ypad,
                        const float* __restrict__ xsq,
                        const float* __restrict__ ysq,
                        float* __restrict__ fg, int updateF) {
  // B fragments in WMMA lane layout: entry [ct*32 + lane]:
  //   lane 0-15 : (x, y)           of column point ct*16+lane
  //   lane 16-31: (z, colTerm)     of column point ct*16+lane-16
  __shared__ v2f    sB[NTILES * 32];          // 32 KB
  __shared__ float4 sRowPts[ROWS_PER_BLOCK];  //  2 KB
  __shared__ float  sRowTerm[ROWS_PER_BLOCK];

  const int p    = blockIdx.x >> 4;         // problem 0..23
  const int tgrp = blockIdx.x & 15;         // row stripe within problem
  const int b    = p / 3;
  const int kind = p - 3 * b;               // 0:xy 1:xx 2:yy

  const float4* aPts = (kind == 2) ? (ypad + b * NPTS) : (xpad + b * NPTS);
  const float*  aSq  = (kind == 2) ? (ysq  + b * NPTS) : (xsq  + b * NPTS);
  const float4* bPts = (kind == 1) ? (xpad + b * NPTS) : (ypad + b * NPTS);
  const float*  bSq  = (kind == 1) ? (xsq  + b * NPTS) : (ysq  + b * NPTS);

  float* f = fg + p * 2 * NPTS;
  float* g = f + NPTS;

  const float4* rowPtsG = updateF ? aPts : bPts;
  const float*  rowSqG  = updateF ? aSq  : bSq;
  const float4* colPtsG = updateF ? bPts : aPts;
  const float*  colSqG  = updateF ? bSq  : aSq;
  const float*  dualIn  = updateF ? g : f;
  float*        dualOut = updateF ? f : g;

  const int rowBlockBase = tgrp * ROWS_PER_BLOCK;

  // ---- Stage B fragments. (x,y) halves copied async when possible; the
  // (z, colTerm) halves are computed. colTerm = (dual_in - 0.5|c|^2)*scale.
#if __has_builtin(__builtin_amdgcn_global_load_async_to_lds_b64) && \
    __has_builtin(__builtin_amdgcn_s_wait_asynccnt)
  for (int j = threadIdx.x; j < NPTS; j += BLK) {
    __builtin_amdgcn_global_load_async_to_lds_b64(
        (gbl_v2i_p)(colPtsG + j),
        (lds_v2i_p)(&sB[(j >> 4) * 32 + (j & 15)]), 0, 0);
  }
  for (int j = threadIdx.x; j < NPTS; j += BLK) {
    v2f t;
    t.x = colPtsG[j].z;
    t.y = (dualIn[j] - 0.5f * colSqG[j]) * SK_SCALE;
    sB[(j >> 4) * 32 + 16 + (j & 15)] = t;
  }
  __builtin_amdgcn_s_wait_asynccnt(0);
#else
  for (int j = threadIdx.x; j < NPTS; j += BLK) {
    float4 pt = colPtsG[j];
    v2f lo, hiv;
    lo.x  = pt.x; lo.y  = pt.y;
    hiv.x = pt.z; hiv.y = (dualIn[j] - 0.5f * colSqG[j]) * SK_SCALE;
    sB[(j >> 4) * 32 + (j & 15)]      = lo;
    sB[(j >> 4) * 32 + 16 + (j & 15)] = hiv;
  }
#endif
  for (int i = threadIdx.x; i < ROWS_PER_BLOCK; i += BLK) {
    sRowPts[i]  = rowPtsG[rowBlockBase + i];
    sRowTerm[i] = -0.5f * rowSqG[rowBlockBase + i] * SK_SCALE;
  }
  __syncthreads();

  const int wave = threadIdx.x >> 5;
  const int lane = threadIdx.x & 31;
  const bool hi  = lane >= 16;
  const int l16  = lane & 15;
  const int rowBase = wave * 16;            // within the 128-row stripe

  // A fragment (16x4 f32): coords pre-scaled by 1/eps*log2e, K=3 slot = 1.0
  // so that (A x B)[i][j] = scale*dot_ij + colTerm_j.
  float4 ap = sRowPts[rowBase + l16];
  v2f afrag;
  afrag.x = (hi ? ap.z : ap.x) * SK_SCALE;
  afrag.y = hi ? 1.0f : ap.y * SK_SCALE;

  float rT[8];
  v8f cRT;                                   // C operand: per-row term
#pragma unroll
  for (int r = 0; r < 8; ++r) {
    rT[r]  = sRowTerm[rowBase + r + (hi ? 8 : 0)];
    cRT[r] = rT[r];
  }

  float m[8];
#pragma unroll
  for (int r = 0; r < 8; ++r) m[r] = -INFINITY;

  // ---- Pass 1: row maxima of logit (log2 units). Loop = ds_load_b64 +
  // WMMA (output already includes row+col terms) + v_max3.
  for (int ct = 0; ct < NTILES; ++ct) {
    v2f bfrag = sB[ct * 32 + lane];
    v8f d = __builtin_amdgcn_wmma_f32_16x16x4_f32(
        false, afrag, false, bfrag, (short)0, cRT, false, false);
#pragma unroll
    for (int r = 0; r < 8; ++r)
      m[r] = fmaxf(m[r], d[r]);
  }

  // Max across the 16 column-slot lanes (all lanes end with the row max).
#pragma unroll
  for (int r = 0; r < 8; ++r)
    for (int xr = 1; xr < 16; xr <<= 1)
      m[r] = fmaxf(m[r], __shfl_xor(m[r], xr));

  // Fold -max into the C operand; pass 2 is exp2-accumulate only.
  v8f cRT2;
  float s[8];
#pragma unroll
  for (int r = 0; r < 8; ++r) {
    cRT2[r] = rT[r] - m[r];
    s[r]    = 0.0f;
  }

  // ---- Pass 2: s = sum_j 2^(logit - m).  WMMA + 1 native v_exp per element.
  for (int ct = 0; ct < NTILES; ++ct) {
    v2f bfrag = sB[ct * 32 + lane];
    v8f d = __builtin_amdgcn_wmma_f32_16x16x4_f32(
        false, afrag, false, bfrag, (short)0, cRT2, false, false);
#pragma unroll
    for (int r = 0; r < 8; ++r)
      s[r] += exp2f(d[r]);
  }

  // Plain add-reduce across the 16 lanes (shared max => no rescaling).
#pragma unroll
  for (int r = 0; r < 8; ++r)
    for (int xr = 1; xr < 16; xr <<= 1)
      s[r] += __shfl_xor(s[r], xr);

  if (l16 == 0) {
#pragma unroll
    for (int r = 0; r < 8; ++r) {
      int row = rowBlockBase + rowBase + r + (hi ? 8 : 0);
      float lse_nat = (m[r] + __log2f(s[r])) * LN2;   // back to natural log
      dualOut[row] = -SK_EPS * (lse_nat - LOG_N);
    }
  }
}

// ---------------------------------------------------------------------------
// Finalize: out = mean_b [ (mean f + mean g)_xy - 0.5*(..)_xx - 0.5*(..)_yy ]
// ---------------------------------------------------------------------------
__global__ void sk_final(const float* __restrict__ fg, float* __restrict__ out) {
  __shared__ float red[BLK];
  float acc = 0.0f;
  for (int idx = threadIdx.x; idx < NPROB * 2 * NPTS; idx += BLK) {
    int p   = idx / (2 * NPTS);
    float w = ((p % 3) == 0) ? 1.0f : -0.5f;
    acc += w * fg[idx];
  }
  red[threadIdx.x] = acc;
  __syncthreads();
  for (int st = BLK / 2; st > 0; st >>= 1) {
    if (threadIdx.x < st) red[threadIdx.x] += red[threadIdx.x + st];
    __syncthreads();
  }
  if (threadIdx.x == 0)
    out[0] = red[0] * (1.0f / (float)(NBATCH * NPTS));
}

// ---------------------------------------------------------------------------
extern "C" void kernel_launch(void* const* d_in, const int* in_sizes, int n_in,
                              void* d_out, int out_size, void* d_ws, size_t ws_size,
                              hipStream_t stream) {
  const float* pcs1 = (const float*)d_in[0];
  const float* pcs2 = (const float*)d_in[1];

  char* ws = (char*)d_ws;
  float4* xpad = (float4*)ws; ws += (size_t)NBATCH * NPTS * sizeof(float4);
  float4* ypad = (float4*)ws; ws += (size_t)NBATCH * NPTS * sizeof(float4);
  float*  xsq  = (float*) ws; ws += (size_t)NBATCH * NPTS * sizeof(float);
  float*  ysq  = (float*) ws; ws += (size_t)NBATCH * NPTS * sizeof(float);
  float*  fg   = (float*) ws; // NPROB * 2 * NPTS floats

  sk_init<<<(NBATCH * NPTS) / BLK, BLK, 0, stream>>>(pcs1, pcs2, xpad, ypad, xsq, ysq);
  (void)hipMemsetAsync(fg, 0, (size_t)NPROB * 2 * NPTS * sizeof(float), stream);

  for (int it = 0; it < SK_ITERS; ++it) {
    sk_pass<<<NPROB * BLOCKS_PER_PROB, BLK, 0, stream>>>(xpad, ypad, xsq, ysq, fg, 1);
    sk_pass<<<NPROB * BLOCKS_PER_PROB, BLK, 0, stream>>>(xpad, ypad, xsq, ysq, fg, 0);
  }

  sk_final<<<1, BLK, 0, stream>>>(fg, (float*)d_out);
}